// IzhikevichNet_41807211659959
// MI455X (gfx1250) — compile-verified
//
#include <hip/hip_runtime.h>

typedef float v2f __attribute__((ext_vector_type(2)));
typedef float v8f __attribute__((ext_vector_type(8)));

#define B_   1024
#define T_   200
#define NIN_ 784
#define H1_  100
#define H2_  50

#define W1S  786   // padded LDS row stride for W1 (floats): (18h+k)%64 distinct -> conflict-free
#define SPS  102   // padded LDS row stride for spk1 (floats): (38m+k)%64 distinct

__launch_bounds__(224, 1)
__global__ void izh_fused_kernel(const float* __restrict__ x,
                                 const float* __restrict__ W1,
                                 const float* __restrict__ b1,
                                 const float* __restrict__ W2,
                                 const float* __restrict__ b2,
                                 float* __restrict__ spk2_rec,
                                 float* __restrict__ mem2_rec,
                                 float* __restrict__ spk1_rec)
{
    __shared__ float w1s[H1_ * W1S];    // 314,400 B
    __shared__ float spk1s[16 * SPS];   //   6,528 B   (total 320,928 B < 320 KB WGP LDS)

    const int tid   = threadIdx.x;
    const int wave  = tid >> 5;     // 0..6, one 16-wide N tile each
    const int lane  = tid & 31;
    const int l16   = lane & 15;
    const int lhalf = lane >> 4;    // fragment half: K/M offset selector
    const int bm0   = blockIdx.x << 4;   // 16 batch rows per workgroup

    // ---- Stage W1 (100 x 784 f32) into padded LDS once; amortized over 200 steps ----
    for (int idx = tid; idx < H1_ * (NIN_ / 2); idx += 224) {
        int h = idx / (NIN_ / 2);
        int j = idx - h * (NIN_ / 2);
        *(float2*)(&w1s[h * W1S + 2 * j]) = *(const float2*)(&W1[h * NIN_ + 2 * j]);
    }
    __syncthreads();

    // Fragment column (N) index for this lane/wave
    const int   hcol = wave * 16 + l16;            // layer-1 neuron col (0..111, valid <100)
    const int   hc   = (hcol < H1_) ? hcol : 0;
    const float b1v  = b1[hc];
    const int   h2c  = (hcol < H2_) ? hcol : 0;    // layer-2 col (waves 0..3, valid <50)
    const float b2v  = b2[h2c];

    // Per-lane streaming bases.
    // A frag (16x4, f32): lane l -> M=l%16, K = vgpr + 2*(l/16)  => b64 load of 2 consecutive k
    const float* xA  = x + (size_t)(bm0 + l16) * T_ * NIN_ + 2 * lhalf;
    // B frag (4x16):      lane l -> N=l%16, K = vgpr + 2*(l/16)  => b64 along k in padded LDS row
    const float* w1B = &w1s[hc * W1S + 2 * lhalf];
    const float* w2B = W2 + h2c * H1_ + 2 * lhalf;
    const float* s1A = &spk1s[l16 * SPS + 2 * lhalf];

    // Wave-spread prefetch offsets: 7 waves x 4 offsets x 28-float spacing covers a
    // full 784-float row width in 128B-line granules.
    const int pf0 = wave * 112;

    // Persistent state in C-fragment layout: element r,lane -> (M = r + 8*lhalf, N = l16)
    v8f v1, u1, Ipre, Ipost, v2;
    #pragma unroll
    for (int r = 0; r < 8; ++r) {
        v1[r] = -65.0f; u1[r] = -13.0f; Ipre[r] = 0.0f; Ipost[r] = 0.0f; v2[r] = -65.0f;
    }

    for (int t = 0; t < T_; ++t) {
        const float* xAt = xA + t * NIN_;

        // Prefetch next timestep's A tile (all addresses in-bounds due to guard)
        if (t + 1 < T_) {
            const float* pfp = xAt + NIN_ + pf0;
            __builtin_prefetch(pfp,      0, 3);
            __builtin_prefetch(pfp + 28, 0, 3);
            __builtin_prefetch(pfp + 56, 0, 3);
            __builtin_prefetch(pfp + 84, 0, 3);
        }

        // ---- Layer-1 GEMM tile: C[16x16] (M=batch, N=this wave's h tile), K=784 ----
        // Two independent accumulator chains to break the WMMA->WMMA RAW dependency.
        v8f acc0, acc1;
        #pragma unroll
        for (int r = 0; r < 8; ++r) { acc0[r] = 0.0f; acc1[r] = 0.0f; }
        #pragma unroll 4
        for (int k = 0; k < NIN_; k += 8) {
            v2f a0 = *(const v2f*)(xAt + k);        // global, WGP$-shared across waves
            v2f b0 = *(const v2f*)(w1B + k);        // LDS, conflict-free
            v2f a1 = *(const v2f*)(xAt + k + 4);
            v2f b1f = *(const v2f*)(w1B + k + 4);
            acc0 = __builtin_amdgcn_wmma_f32_16x16x4_f32(false, a0, false, b0,
                                                         (short)0, acc0, false, false);
            acc1 = __builtin_amdgcn_wmma_f32_16x16x4_f32(false, a1, false, b1f,
                                                         (short)0, acc1, false, false);
        }

        // ---- Izhikevich dynamics, elementwise on the fragment ----
        v8f spkf;
        #pragma unroll
        for (int r = 0; r < 8; ++r) {
            float cur   = (acc0[r] + acc1[r]) + b1v;
            float ipre  = 0.95f * Ipre[r] + cur;
            float ipost = 0.90f * Ipost[r] + cur;
            float I     = ipre - ipost;
            float v     = v1[r];
            float u     = u1[r];
            float vn    = v + (0.04f * v * v + 5.0f * v + 140.0f - u + I);
            float un    = u + 0.02f * (0.2f * v - u);
            float s     = (vn >= 30.0f) ? 1.0f : 0.0f;
            vn          = (s > 0.0f) ? -65.0f : vn;
            un         += s * 8.0f;
            Ipre[r] = ipre; Ipost[r] = ipost; v1[r] = vn; u1[r] = un; spkf[r] = s;
        }

        // Publish spikes: LDS (for layer-2 A matrix) + global record
        if (hcol < H1_) {
            #pragma unroll
            for (int r = 0; r < 8; ++r) {
                int m = r + 8 * lhalf;
                spk1s[m * SPS + hcol] = spkf[r];
                spk1_rec[((size_t)t * B_ + bm0 + m) * H1_ + hcol] = spkf[r];
            }
        }
        __syncthreads();

        // ---- Layer 2: cur2 = spk1 @ W2^T + b2; leaky integrate (waves 0..3) ----
        if (wave < 4) {
            v8f acc2;
            #pragma unroll
            for (int r = 0; r < 8; ++r) acc2[r] = 0.95f * v2[r] + b2v;  // beta*v2 + b2 as C
            #pragma unroll 5
            for (int k = 0; k < H1_; k += 4) {
                v2f a = *(const v2f*)(s1A + k);   // spk1 from LDS
                v2f b = *(const v2f*)(w2B + k);   // W2 (20 KB, cache-resident)
                acc2 = __builtin_amdgcn_wmma_f32_16x16x4_f32(false, a, false, b,
                                                             (short)0, acc2, false, false);
            }
            #pragma unroll
            for (int r = 0; r < 8; ++r) {
                float vn2 = acc2[r];
                float s2  = (vn2 > 1.0f) ? 1.0f : 0.0f;
                vn2      -= s2;                   // subtract-reset
                v2[r]     = vn2;
                if (hcol < H2_) {
                    int m = r + 8 * lhalf;
                    size_t o = ((size_t)t * B_ + bm0 + m) * H2_ + hcol;
                    spk2_rec[o] = s2;
                    mem2_rec[o] = vn2;
                }
            }
        }
        __syncthreads();   // protect spk1s against next timestep's writers
    }
}

extern "C" void kernel_launch(void* const* d_in, const int* in_sizes, int n_in,
                              void* d_out, int out_size, void* d_ws, size_t ws_size,
                              hipStream_t stream) {
    (void)in_sizes; (void)n_in; (void)out_size; (void)d_ws; (void)ws_size;
    const float* x  = (const float*)d_in[0];
    const float* W1 = (const float*)d_in[1];
    const float* b1 = (const float*)d_in[2];
    const float* W2 = (const float*)d_in[3];
    const float* b2 = (const float*)d_in[4];

    float* out  = (float*)d_out;
    float* spk2 = out;                                  // [200,1024,50]
    float* mem2 = spk2 + (size_t)T_ * B_ * H2_;         // [200,1024,50]
    float* spk1 = mem2 + (size_t)T_ * B_ * H2_;         // [200,1024,100]

    dim3 grid(B_ / 16);
    dim3 block(224);   // 7 wave32 waves: one per 16-wide N tile of H1
    hipLaunchKernelGGL(izh_fused_kernel, grid, block, 0, stream,
                       x, W1, b1, W2, b2, spk2, mem2, spk1);
}